// GCNLayer_69492570849698
// MI455X (gfx1250) — compile-verified
//
#include <hip/hip_runtime.h>
#include <hip/hip_bf16.h>

// ---------------------------------------------------------------------------
// GCN layer for MI455X (gfx1250, wave32):
//   h = x W^T + b                         (fp32 WMMA 16x16x4, A double-buffered)
//   deg-normalized edge aggregation       (L2-resident gather + fp32 atomics)
//   residual + LayerNorm + ReLU + mask    (wave32 shfl reductions)
// ---------------------------------------------------------------------------

typedef float v2f __attribute__((ext_vector_type(2)));
typedef float v8f __attribute__((ext_vector_type(8)));

#define NB      8
#define NN      4500
#define NE      72000
#define DIM     128            // IN_DIM == OUT_DIM
#define TOTAL   (NB * NN)      // 36000 nodes
#define TOTE    (NB * NE)      // 576000 edges
#define EPS     1e-5f

// ---------------------------------------------------------------------------
// Kernel 1: h[36000,128] = x[36000,128] @ W^T + b  via V_WMMA_F32_16X16X4_F32
// One wave computes a 16-row M-tile across all 128 output columns (8 v8f accs).
//
// LDS layout (pair-interleaved B-matrix order) so each B operand (rows k,k+1
// at column n) is one contiguous 8B element -> single ds_load_b64 into an
// even-aligned VGPR pair, no repacking movs:
//     Wlds[(k>>1)*256 + n*2 + (k&1)] = W[n*DIM + k]
//
// A fragment is explicitly double-buffered: the k+4 global_load_b64 issues
// before the 8-WMMA burst for k, hiding VMEM latency inside the XDL burst.
// ---------------------------------------------------------------------------
__global__ __launch_bounds__(256) void k_linear_wmma(
    const float* __restrict__ x, const float* __restrict__ W,
    const float* __restrict__ bias, float* __restrict__ h)
{
    __shared__ float Wlds[DIM * DIM];     // 64 KB of the 320 KB WGP LDS

    const int tid = threadIdx.x;
    // cooperative pair-interleaved load of W (row-major [OUT][IN])
    for (int i = tid; i < DIM * DIM; i += 256) {
        const int n = i >> 7;
        const int k = i & 127;
        Wlds[(k >> 1) * 256 + n * 2 + (k & 1)] = W[n * DIM + k];
    }
    __syncthreads();

    const int wave = tid >> 5;                       // 8 waves / block
    const int lane = tid & 31;
    const int m0   = (blockIdx.x * 8 + wave) * 16;   // 16-row tile base
    if (m0 >= TOTAL) return;                         // wave-uniform

    const int half = lane >> 4;                      // 0 | 1 (K pair select)
    const int l16  = lane & 15;

    v8f c[8];
    const v8f zero = {0.f, 0.f, 0.f, 0.f, 0.f, 0.f, 0.f, 0.f};
#pragma unroll
    for (int nt = 0; nt < 8; ++nt) c[nt] = zero;

    // A-matrix: lane l16 = row M; VGPR0/1 hold K = kk+2*half, kk+2*half+1
    const float* arow = x + (size_t)(m0 + l16) * DIM + 2 * half;
    __builtin_prefetch(arow, 0, 3);                  // global_prefetch_b8

    // B base for this lane: pair row (kk>>1)+half, column l16
    const v2f* bl = (const v2f*)&Wlds[half * 256 + l16 * 2];

    v2f a_cur = *(const v2f*)(arow);                 // kk = 0 fragment
    for (int kk = 0; kk < DIM; kk += 4) {
        v2f a_next;
        if (kk + 4 < DIM)                            // issue next A load early
            a_next = *(const v2f*)(arow + kk + 4);
        const v2f* bp = bl + (size_t)(kk >> 1) * 128;  // 128 v2f per pair-row
#pragma unroll
        for (int nt = 0; nt < 8; ++nt) {
            const v2f bm = bp[nt * 16];              // ds_load_b64, imm offset
            c[nt] = __builtin_amdgcn_wmma_f32_16x16x4_f32(
                false, a_cur, false, bm, (short)0, c[nt], false, false);
        }
        a_cur = a_next;
    }

    // D layout: VGPR v -> row v + 8*half, col = l16 (+16*nt)
#pragma unroll
    for (int nt = 0; nt < 8; ++nt) {
        const float bv = bias[nt * 16 + l16];
#pragma unroll
        for (int v = 0; v < 8; ++v) {
            const int r = v + 8 * half;
            h[(size_t)(m0 + r) * DIM + nt * 16 + l16] = c[nt][v] + bv;
        }
    }
}

// ---------------------------------------------------------------------------
// Kernel 2a: deg = 1.0 (self loop)
// ---------------------------------------------------------------------------
__global__ __launch_bounds__(256) void k_init_deg(float* __restrict__ deg)
{
    const int i = blockIdx.x * blockDim.x + threadIdx.x;
    if (i < TOTAL) deg[i] = 1.0f;
}

// ---------------------------------------------------------------------------
// Kernel 2b: deg[tgt] += edge_mask  (valid-edge in-degree)
// edge_index flat layout: [B, 2, E]
// ---------------------------------------------------------------------------
__global__ __launch_bounds__(256) void k_degree(
    const int* __restrict__ ei, const float* __restrict__ em,
    float* __restrict__ deg)
{
    const int e = blockIdx.x * blockDim.x + threadIdx.x;
    if (e >= TOTE) return;
    const int b = e / NE;
    const int r = e - b * NE;
    const float w = em[e];
    if (w == 0.0f) return;
    const int tgt = ei[(size_t)b * 2 * NE + NE + r] + b * NN;
    atomicAdd(&deg[tgt], w);
}

// ---------------------------------------------------------------------------
// Kernel 3: dinv = rsqrt(deg) (in-place), agg = h * dinv (self contribution)
// One 128-thread block per node.
// ---------------------------------------------------------------------------
__global__ __launch_bounds__(128) void k_selfagg(
    const float* __restrict__ h, float* __restrict__ deg,
    float* __restrict__ agg)
{
    const int n = blockIdx.x;
    const float dv = rsqrtf(deg[n]);
    __syncthreads();                       // all reads of deg[n] done
    if (threadIdx.x == 0) deg[n] = dv;     // deg becomes dinv
    const int d = threadIdx.x;
    agg[(size_t)n * DIM + d] = h[(size_t)n * DIM + d] * dv;
}

// ---------------------------------------------------------------------------
// Kernel 4: agg[tgt] += h[src] * (dinv[src]*dinv[tgt]*ew)
// One wave32 per edge; float4 gather (L2-resident), 4 fp32 atomics per lane.
// ---------------------------------------------------------------------------
__global__ __launch_bounds__(256) void k_scatter(
    const int* __restrict__ ei, const float* __restrict__ em,
    const float* __restrict__ dinv, const float* __restrict__ h,
    float* __restrict__ agg)
{
    const int gtid = blockIdx.x * blockDim.x + threadIdx.x;
    const int e    = gtid >> 5;
    const int lane = threadIdx.x & 31;
    if (e >= TOTE) return;

    const float w = em[e];
    if (w == 0.0f) return;

    const int b = e / NE;
    const int r = e - b * NE;
    const int* eib = ei + (size_t)b * 2 * NE;
    const int s = eib[r]      + b * NN;
    const int t = eib[NE + r] + b * NN;

    const float coef = dinv[s] * dinv[t] * w;

    const float4 v = ((const float4*)(h + (size_t)s * DIM))[lane];
    float* at = agg + (size_t)t * DIM + lane * 4;
    atomicAdd(at + 0, v.x * coef);
    atomicAdd(at + 1, v.y * coef);
    atomicAdd(at + 2, v.z * coef);
    atomicAdd(at + 3, v.w * coef);
}

// ---------------------------------------------------------------------------
// Kernel 5: out = relu(LN(h + agg) * gamma + beta) * node_mask
// One wave32 per node, 4 elements per lane, shfl_xor tree reduction.
// ---------------------------------------------------------------------------
__global__ __launch_bounds__(256) void k_norm(
    const float* __restrict__ h, const float* __restrict__ agg,
    const float* __restrict__ gamma, const float* __restrict__ beta,
    const float* __restrict__ nmask, float* __restrict__ out)
{
    const int gtid = blockIdx.x * blockDim.x + threadIdx.x;
    const int n    = gtid >> 5;
    const int lane = threadIdx.x & 31;
    if (n >= TOTAL) return;

    const float4 hv = ((const float4*)(h   + (size_t)n * DIM))[lane];
    const float4 av = ((const float4*)(agg + (size_t)n * DIM))[lane];
    float4 v;
    v.x = hv.x + av.x; v.y = hv.y + av.y;
    v.z = hv.z + av.z; v.w = hv.w + av.w;

    float s  = v.x + v.y + v.z + v.w;
    float sq = v.x * v.x + v.y * v.y + v.z * v.z + v.w * v.w;
#pragma unroll
    for (int m = 16; m > 0; m >>= 1) {
        s  += __shfl_xor(s,  m, 32);
        sq += __shfl_xor(sq, m, 32);
    }
    const float mu  = s * (1.0f / DIM);
    const float var = sq * (1.0f / DIM) - mu * mu;
    const float inv = rsqrtf(var + EPS);
    const float mk  = nmask[n];

    const float4 g = ((const float4*)gamma)[lane];
    const float4 b = ((const float4*)beta)[lane];
    float4 o;
    o.x = fmaxf((v.x - mu) * inv * g.x + b.x, 0.0f) * mk;
    o.y = fmaxf((v.y - mu) * inv * g.y + b.y, 0.0f) * mk;
    o.z = fmaxf((v.z - mu) * inv * g.z + b.z, 0.0f) * mk;
    o.w = fmaxf((v.w - mu) * inv * g.w + b.w, 0.0f) * mk;
    ((float4*)out)[(size_t)n * (DIM / 4) + lane] = o;
}

// ---------------------------------------------------------------------------
// Launch
// ---------------------------------------------------------------------------
extern "C" void kernel_launch(void* const* d_in, const int* in_sizes, int n_in,
                              void* d_out, int out_size, void* d_ws, size_t ws_size,
                              hipStream_t stream)
{
    const float* x     = (const float*)d_in[0];   // [B,N,128]
    const int*   ei    = (const int*)  d_in[1];   // [B,2,E]
    const float* nmask = (const float*)d_in[2];   // [B,N]
    const float* emask = (const float*)d_in[3];   // [B,E]
    const float* W     = (const float*)d_in[4];   // [128,128]
    const float* bias  = (const float*)d_in[5];   // [128]
    const float* gamma = (const float*)d_in[6];   // [128]
    const float* beta  = (const float*)d_in[7];   // [128]
    float* out = (float*)d_out;

    // workspace carve-out: h | agg | deg(->dinv)
    float* h   = (float*)d_ws;
    float* agg = h   + (size_t)TOTAL * DIM;
    float* deg = agg + (size_t)TOTAL * DIM;

    // 1) WMMA linear: 2250 M-tiles, 8 waves (tiles) per 256-thread block
    k_linear_wmma<<<(TOTAL / 16 + 7) / 8, 256, 0, stream>>>(x, W, bias, h);

    // 2) degree with self loop
    k_init_deg<<<(TOTAL + 255) / 256, 256, 0, stream>>>(deg);
    k_degree<<<(TOTE + 255) / 256, 256, 0, stream>>>(ei, emask, deg);

    // 3) dinv + self contribution (initializes agg)
    k_selfagg<<<TOTAL, 128, 0, stream>>>(h, deg, agg);

    // 4) edge scatter: one wave per edge -> 8 edges per block
    k_scatter<<<TOTE / 8, 256, 0, stream>>>(ei, emask, deg, h, agg);

    // 5) residual + LayerNorm + ReLU + mask: 8 nodes per block
    k_norm<<<(TOTAL + 7) / 8, 256, 0, stream>>>(h, agg, gamma, beta, nmask, out);
}